// InfoNCE_15539191677091
// MI455X (gfx1250) — compile-verified
//
#include <hip/hip_runtime.h>
#include <hip/hip_bf16.h>
#include <math.h>

typedef float v2f __attribute__((ext_vector_type(2)));
typedef float v8f __attribute__((ext_vector_type(8)));

#define N_SAMP   1024
#define IN_DIM   64
#define HIDDEN   128
#define HP       132      // padded LDS row stride (floats): 16B-aligned rows

__device__ __forceinline__ float softplusf(float x) {
    return (x > 20.0f) ? x : log1pf(expf(x));
}

__device__ __forceinline__ void lse_update(float v, float& m, float& s) {
    if (v > m) { s = s * expf(m - v) + 1.0f; m = v; }
    else       { s += expf(v - m); }
}

// gfx1250 async direct-to-LDS copy (ASYNCcnt-tracked), one 16B packet per lane.
__device__ __forceinline__ void async_ld_b128(const float* gsrc, float* ldst) {
    const unsigned int laddr = (unsigned int)(uintptr_t)ldst;  // low 32 bits = LDS offset
    asm volatile("global_load_async_to_lds_b128 %0, %1, off"
                 :: "v"(laddr), "v"(gsrc) : "memory");
}
__device__ __forceinline__ void wait_async0() {
    asm volatile("s_wait_asynccnt 0" ::: "memory");
}

// ---------------------------------------------------------------------------
// Kernel 1: fp32 WMMA GEMM  Out[n,h] = sum_k A[n,k]*W[k,h] (+ optional bias[h])
// One wave per 16x16 tile, 16x V_WMMA_F32_16X16X4_F32 fully unrolled (K=64).
// ---------------------------------------------------------------------------
__global__ __launch_bounds__(32) void gemm_wmma_f32_kernel(
    const float* __restrict__ A,      // Nrows x 64
    const float* __restrict__ W,      // 64 x 128
    const float* __restrict__ bias,   // 128 or nullptr
    float* __restrict__ Out)          // Nrows x 128
{
    const int lane = threadIdx.x;
    const int tm   = blockIdx.x;
    const int tn   = blockIdx.y;
    const int mrow = lane & 15;
    const int kh   = lane >> 4;
    const int col  = tn * 16 + (lane & 15);

    v8f acc = {};
#pragma unroll
    for (int k0 = 0; k0 < IN_DIM; k0 += 4) {
        const int ka = k0 + 2 * kh;
        v2f a, b;
        a.x = A[(tm * 16 + mrow) * IN_DIM + ka];
        a.y = A[(tm * 16 + mrow) * IN_DIM + ka + 1];
        b.x = W[(ka    ) * HIDDEN + col];
        b.y = W[(ka + 1) * HIDDEN + col];
        acc = __builtin_amdgcn_wmma_f32_16x16x4_f32(
                  false, a, false, b, (short)0, acc, false, false);
    }

    float bv = 0.0f;
    if (bias) bv = bias[col];

    const int rbase = 8 * kh;
#pragma unroll
    for (int r = 0; r < 8; ++r) {
        Out[(tm * 16 + rbase + r) * HIDDEN + col] = acc[r] + bv;
    }
}

// ---------------------------------------------------------------------------
// Kernel 2: pairwise T1[i,j] = softplus(dot(relu(yp[i]+xpb[j]), W2) + b2),
// streaming logsumexp over j, plus diagonal T0[i] = T1[i,i].
// 64 WGs x 256 threads; WG owns 16 i-rows. 64-j tiles are double-buffered in
// LDS via global_load_async_to_lds_b128 (load of tile jt+1 overlaps compute
// of tile jt). 2i x 2j register tiling: thread t handles i in {t>>5, (t>>5)+8}
// and j in {t&31, (t&31)+32} -> 4 accumulators, 5 LDS-128 reads / 48 VALU ops.
// ---------------------------------------------------------------------------
#define STEP(Y, X, W, ACC)                                   \
    ACC = fmaf(fmaxf((Y).x + (X).x, 0.0f), (W).x, ACC);      \
    ACC = fmaf(fmaxf((Y).y + (X).y, 0.0f), (W).y, ACC);      \
    ACC = fmaf(fmaxf((Y).z + (X).z, 0.0f), (W).z, ACC);      \
    ACC = fmaf(fmaxf((Y).w + (X).w, 0.0f), (W).w, ACC);

__global__ __launch_bounds__(256) void pairwise_lse_kernel(
    const float* __restrict__ xpb,   // 1024 x 128 (x@Wx + b1)
    const float* __restrict__ yp,    // 1024 x 128 (y@Wy)
    const float* __restrict__ W2,    // 128
    const float* __restrict__ b2p,   // 1
    float* __restrict__ partial)     // 64
{
    __shared__ __align__(16) float  sX0[64 * HP];
    __shared__ __align__(16) float  sX1[64 * HP];
    __shared__ __align__(16) float  sYp[16 * HP];
    __shared__ float4 sW2[32];
    __shared__ float  smM[16][32];
    __shared__ float  smS[16][32];
    __shared__ float  wres[16];

    const int t  = threadIdx.x;
    const int wg = blockIdx.x;       // 0..63
    const int i0 = t >> 5;           // 0..7  (thread's i rows: i0, i0+8)
    const int lj = t & 31;           // 0..31 (thread's j cols: lj, lj+32)

    // Prologue: kick off async load of j-tile 0, stage yp tile + W2 meanwhile.
    {
#pragma unroll
        for (int k = 0; k < 8; ++k) {
            const int idx = t + k * 256;           // 0..2047 float4 packets
            const int r = idx >> 5, c = idx & 31;
            async_ld_b128(xpb + (0 * 64 + r) * HIDDEN + c * 4, &sX0[r * HP + c * 4]);
        }
        for (int idx = t; idx < 16 * 32; idx += 256) {
            const int r = idx >> 5, c = idx & 31;
            const float4 v = ((const float4*)(yp + (wg * 16 + r) * HIDDEN))[c];
            *(float4*)&sYp[r * HP + c * 4] = v;
        }
        if (t < 32) sW2[t] = ((const float4*)W2)[t];
    }
    const float b2 = b2p[0];

    const float4* yr0 = (const float4*)&sYp[i0 * HP];
    const float4* yr1 = (const float4*)&sYp[(i0 + 8) * HP];

    float m0 = -1e30f, s0 = 0.0f;    // lse state for row i0
    float m1 = -1e30f, s1 = 0.0f;    // lse state for row i0+8

    wait_async0();
    __syncthreads();

    for (int jt = 0; jt < 16; ++jt) {
        float* cur = (jt & 1) ? sX1 : sX0;
        float* nxt = (jt & 1) ? sX0 : sX1;

        if (jt < 15) {
#pragma unroll
            for (int k = 0; k < 8; ++k) {
                const int idx = t + k * 256;
                const int r = idx >> 5, c = idx & 31;
                async_ld_b128(xpb + ((jt + 1) * 64 + r) * HIDDEN + c * 4,
                              &nxt[r * HP + c * 4]);
            }
        }

        const float4* xr0 = (const float4*)&cur[lj * HP];
        const float4* xr1 = (const float4*)&cur[(lj + 32) * HP];
        float a00 = 0.0f, a01 = 0.0f, a10 = 0.0f, a11 = 0.0f;
#pragma unroll 8
        for (int h4 = 0; h4 < 32; ++h4) {
            const float4 w  = sW2[h4];
            const float4 x0 = xr0[h4];
            const float4 x1 = xr1[h4];
            const float4 y0 = yr0[h4];
            const float4 y1 = yr1[h4];
            STEP(y0, x0, w, a00); STEP(y0, x1, w, a01);
            STEP(y1, x0, w, a10); STEP(y1, x1, w, a11);
        }
        lse_update(softplusf(a00 + b2), m0, s0);
        lse_update(softplusf(a01 + b2), m0, s0);
        lse_update(softplusf(a10 + b2), m1, s1);
        lse_update(softplusf(a11 + b2), m1, s1);

        wait_async0();       // next tile landed (hidden behind compute)
        __syncthreads();     // everyone done with `cur`, `nxt` visible
    }

    smM[i0][lj] = m0;     smS[i0][lj] = s0;
    smM[i0 + 8][lj] = m1; smS[i0 + 8][lj] = s1;
    __syncthreads();

    if (t < 16) {
        const int i_local = t;
        float M = -1e30f;
#pragma unroll
        for (int k = 0; k < 32; ++k) M = fmaxf(M, smM[i_local][k]);
        float S = 0.0f;
#pragma unroll
        for (int k = 0; k < 32; ++k) S += smS[i_local][k] * expf(smM[i_local][k] - M);
        const float lse = M + logf(S);

        // Diagonal term T0[i] = T1[i,i] (b1 folded into xpb).
        const int iglob = wg * 16 + i_local;
        const float4* xr = (const float4*)(xpb + iglob * HIDDEN);
        const float4* yr = (const float4*)&sYp[i_local * HP];
        float acc = 0.0f;
#pragma unroll
        for (int h4 = 0; h4 < 32; ++h4) {
            const float4 w = sW2[h4];
            const float4 xv = xr[h4];
            const float4 yv = yr[h4];
            STEP(yv, xv, w, acc);
        }
        wres[i_local] = softplusf(acc + b2) - lse;
    }
    __syncthreads();

    if (t == 0) {
        float p = 0.0f;
#pragma unroll
        for (int k = 0; k < 16; ++k) p += wres[k];
        partial[wg] = p;
    }
}

// ---------------------------------------------------------------------------
// Kernel 3: deterministic final reduction: mean(T0 - lse) + log(N)
// ---------------------------------------------------------------------------
__global__ __launch_bounds__(64) void finalize_kernel(
    const float* __restrict__ partial, float* __restrict__ out)
{
    __shared__ float sm[64];
    const int t = threadIdx.x;
    sm[t] = partial[t];
    __syncthreads();
    if (t == 0) {
        float s = 0.0f;
        for (int k = 0; k < 64; ++k) s += sm[k];
        out[0] = s / (float)N_SAMP + logf((float)N_SAMP);
    }
}

extern "C" void kernel_launch(void* const* d_in, const int* in_sizes, int n_in,
                              void* d_out, int out_size, void* d_ws, size_t ws_size,
                              hipStream_t stream) {
    const float* x  = (const float*)d_in[0];   // 1024 x 64
    const float* y  = (const float*)d_in[1];   // 1024 x 64
    const float* W1 = (const float*)d_in[2];   // 128 x 128 (rows 0..63=Wx, 64..127=Wy)
    const float* b1 = (const float*)d_in[3];   // 128
    const float* W2 = (const float*)d_in[4];   // 128
    const float* b2 = (const float*)d_in[5];   // 1

    float* xpb     = (float*)d_ws;                 // 1024*128 (x@Wx + b1)
    float* yp      = xpb + N_SAMP * HIDDEN;        // 1024*128 (y@Wy)
    float* partial = yp  + N_SAMP * HIDDEN;        // 64

    dim3 gg(N_SAMP / 16, HIDDEN / 16);
    gemm_wmma_f32_kernel<<<gg, 32, 0, stream>>>(x, W1,                   b1,      xpb);
    gemm_wmma_f32_kernel<<<gg, 32, 0, stream>>>(y, W1 + IN_DIM * HIDDEN, nullptr, yp);

    pairwise_lse_kernel<<<64, 256, 0, stream>>>(xpb, yp, W2, b2, partial);
    finalize_kernel<<<1, 64, 0, stream>>>(partial, (float*)d_out);
}